// DK_2078764172150
// MI455X (gfx1250) — compile-verified
//
#include <hip/hip_runtime.h>
#include <hip/hip_bf16.h>

typedef __attribute__((ext_vector_type(2))) float v2f;
typedef __attribute__((ext_vector_type(8))) float v8f;

#define DEGR 0.017453292519943295f  // np.float32(pi/180)

// Constant-folded 4x4 affine groups of the kinematic chain (row-major).
// re = C0*Rz(a0)*C1*Rz(a1)*C2*Rz(a2)*C3*Rz(a3)*C4*Rz(a4)*C5*Rz(a5)*C6*Rz(a6)*C7
// a = {th0, th1, -th2, -th3, -th4/2, th5/4.5+10, 20+(th6+180)/4.5} degrees.
// C7 columns (2,3) give the initial state: dir=(1,0,0,0), pt=(6,0,0,1).
__device__ __constant__ float CM[7][16] = {
    // C0 = T(0,5,19.5)*Rz(90)
    { 0.f,-1.f, 0.f, 0.f,   1.f, 0.f, 0.f, 5.f,   0.f, 0.f, 1.f,19.5f,  0.f,0.f,0.f,1.f },
    // C1 = T(0,-1.5,2.5)*Ry(90)
    { 0.f, 0.f, 1.f, 0.f,   0.f, 1.f, 0.f,-1.5f, -1.f, 0.f, 0.f, 2.5f,  0.f,0.f,0.f,1.f },
    // C2 = T(3,0,9.5)*Rx(-90)
    { 1.f, 0.f, 0.f, 3.f,   0.f, 0.f, 1.f, 0.f,   0.f,-1.f, 0.f, 9.5f,  0.f,0.f,0.f,1.f },
    // C3 = T(17.5,0,0)*Rx(90)*Rz(180)
    {-1.f, 0.f, 0.f,17.5f,  0.f, 0.f,-1.f, 0.f,   0.f,-1.f, 0.f, 0.f,   0.f,0.f,0.f,1.f },
    // C4 = T(10,0,0)*Ry(90)
    { 0.f, 0.f, 1.f,10.f,   0.f, 1.f, 0.f, 0.f,  -1.f, 0.f, 0.f, 0.f,   0.f,0.f,0.f,1.f },
    // C5 = T(0,0,10)*Rx(-90)*Rz(-90)
    { 0.f, 1.f, 0.f, 0.f,   0.f, 0.f, 1.f, 0.f,   1.f, 0.f, 0.f,10.f,   0.f,0.f,0.f,1.f },
    // C6 = T(0,-1,0)*T(6,0,0)
    { 1.f, 0.f, 0.f, 6.f,   0.f, 1.f, 0.f,-1.f,   0.f, 0.f, 1.f, 0.f,   0.f,0.f,0.f,1.f },
};

// Branch-free sin/cos for |x| <= ~1.6 rad: half-angle + Cephes pi/4-range minimax.
// Error ~1e-8 relative (sh,ch), ~4e-8 absolute after doubling -- fp32 ulp level.
__device__ __forceinline__ void fast_sincos(float x, float& s, float& c) {
    const float h = 0.5f * x;
    const float z = h * h;
    const float ps = __builtin_fmaf(__builtin_fmaf(-1.9515295891e-4f, z,
                                     8.3321608736e-3f), z, -1.6666654611e-1f);
    const float sh = __builtin_fmaf(h * z, ps, h);
    const float pc = __builtin_fmaf(__builtin_fmaf(2.443315711809948e-5f, z,
                                     -1.388731625493765e-3f), z, 4.166664568298827e-2f);
    const float ch = __builtin_fmaf(z * z, pc, __builtin_fmaf(-0.5f, z, 1.0f));
    s = 2.0f * sh * ch;                          // sin(x) = 2 sin(h) cos(h)
    c = __builtin_fmaf(-2.0f * sh, sh, 1.0f);    // cos(x) = 1 - 2 sin^2(h)
}

// One wave = 16 elements, two parallel chains (direction cols / point cols).
// B operand (4x16 f32, 2 VGPRs): vgpr0={K0|K2}, vgpr1={K1|K3}; column n = lane&15 = element.
// A operand (16x4 f32, 2 VGPRs): A_v[lane] = K[lane&3][2*(lane>>4)+v] -> K replicated 4x,
// so D's duplicate block-rows land exactly in next-B layout (repack = 2 cndmasks).
__global__ __launch_bounds__(256) void fk_wmma_kernel(const float* __restrict__ thetas,
                                                      float* __restrict__ out, int B) {
    const int lane = threadIdx.x & 31;
    const int wave = threadIdx.x >> 5;
    const int col  = lane & 15;               // element slot; lanes n and n+16 share element
    const bool lo  = lane < 16;               // lanes 0-15: rows 0/1; lanes 16-31: rows 2/3
    const int elem = (blockIdx.x * 8 + wave) * 16 + col;
    const int eL   = (elem < B) ? elem : (B - 1);   // clamp: EXEC stays all-1s for WMMA

    // Preload the per-lane A-operand values for all 7 constant stages (stay in VGPRs).
    const int abase = 4 * (lane & 3) + 2 * (lane >> 4);
    float a0[7], a1[7];
#pragma unroll
    for (int k = 0; k < 7; ++k) { a0[k] = CM[k][abase]; a1[k] = CM[k][abase + 1]; }

    const float* t = thetas + (size_t)eL * 7;
    const float t0 = t[0], t1 = t[1], t2 = t[2], t3 = t[3], t4 = t[4], t5 = t[5], t6 = t[6];

    float ang[7];
    ang[0] = t0 * DEGR;
    ang[1] = t1 * DEGR;
    ang[2] = (-t2) * DEGR;
    ang[3] = (-t3) * DEGR;
    ang[4] = (-t4 / 2.0f) * DEGR;
    ang[5] = (t5 / 4.5f + 10.0f) * DEGR;
    ang[6] = (20.0f + (t6 + 180.0f) / 4.5f) * DEGR;

    float cs[7], sn[7];
#pragma unroll
    for (int k = 0; k < 7; ++k) fast_sincos(ang[k], sn[k], cs[k]);

    // Initial state X = C7[:, (2,3)]: dir chain (d) and point chain (p) in B layout.
    float d0 = lo ? 1.0f : 0.0f;              // dir : row0 | row2
    float d1 = 0.0f;                          // dir : row1 | row3
    float p0 = lo ? 6.0f : 0.0f;              // pt  : row0 | row2
    float p1 = lo ? 0.0f : 1.0f;              // pt  : row1 | row3 (w=1)

    const v8f zero8 = {};

#pragma unroll
    for (int k = 6; k >= 0; --k) {
        // Rz(a_k) rotates rows 0/1 only (lanes < 16); identity on rows 2/3.
        const float cl = lo ? cs[k] : 1.0f;
        const float sl = lo ? sn[k] : 0.0f;
        const float rd0 = cl * d0 - sl * d1;
        const float rd1 = sl * d0 + cl * d1;
        const float rp0 = cl * p0 - sl * p1;
        const float rp1 = sl * p0 + cl * p1;

        v2f A;  A.x = a0[k];  A.y = a1[k];    // constant K, replicated 4x
        v2f Bd; Bd.x = rd0;   Bd.y = rd1;
        v2f Bp; Bp.x = rp0;   Bp.y = rp1;

        const v8f dd = __builtin_amdgcn_wmma_f32_16x16x4_f32(
            false, A, false, Bd, (short)0, zero8, false, false);
        const v8f dp = __builtin_amdgcn_wmma_f32_16x16x4_f32(
            false, A, false, Bp, (short)0, zero8, false, false);

        d0 = lo ? dd[0] : dd[2];  d1 = lo ? dd[1] : dd[3];
        p0 = lo ? dp[0] : dp[2];  p1 = lo ? dp[1] : dp[3];
    }

    if (elem < B) {
        float* pt = out + (size_t)3 * (size_t)elem;                       // points region
        float* vc = out + (size_t)3 * (size_t)B + (size_t)3 * (size_t)elem; // vectors region
        if (lo) {
            __builtin_nontemporal_store(p0, pt + 0);
            __builtin_nontemporal_store(p1, pt + 1);
            __builtin_nontemporal_store(d0, vc + 0);
            __builtin_nontemporal_store(d1, vc + 1);
        } else {
            __builtin_nontemporal_store(p0, pt + 2);   // row 2 (row 3 discarded)
            __builtin_nontemporal_store(d0, vc + 2);
        }
    }
}

extern "C" void kernel_launch(void* const* d_in, const int* in_sizes, int n_in,
                              void* d_out, int out_size, void* d_ws, size_t ws_size,
                              hipStream_t stream) {
    (void)n_in; (void)out_size; (void)d_ws; (void)ws_size;
    const float* thetas = (const float*)d_in[0];
    float* out = (float*)d_out;
    const int B = in_sizes[0] / 7;              // 1,048,576
    const int blocks = (B + 127) / 128;         // 128 elements per 256-thread block (8 waves x 16)
    fk_wmma_kernel<<<blocks, 256, 0, stream>>>(thetas, out, B);
}